// GNN_Nodes_49469433315362
// MI455X (gfx1250) — compile-verified
//
#include <hip/hip_runtime.h>
#include <hip/hip_bf16.h>

typedef float v2f __attribute__((ext_vector_type(2)));
typedef float v8f __attribute__((ext_vector_type(8)));

// ---------------------------------------------------------------------------
// Dense GEMM via V_WMMA_F32_16X16X4_F32 (wave32).
// X: [M x 128] row-major.  W: [128 x NOUT] row-major (contiguous).
// out: [M x NOUT].  ACC: false = overwrite, true = out += X@W.
// Block = 256 threads = 8 waves; each wave computes a 16-row slab across all
// NOUT columns. W staged in LDS once per block (<= 64 KB) using the CDNA5
// async global->LDS DMA path (ASYNCcnt-tracked, no VGPR round trip).
// ---------------------------------------------------------------------------
template <int NOUT, bool ACC>
__global__ __launch_bounds__(256) void gemm128_wmma(
    const float* __restrict__ X, const float* __restrict__ W,
    float* __restrict__ out, int M) {
  constexpr int K = 128;
  constexpr int NT = NOUT / 16;
  __shared__ float Wl[K * NOUT];

  {  // async staging: each lane DMAs 16B/iter straight into LDS.
    constexpr int total = K * NOUT / 4;  // float4 elements (multiple of 256)
#pragma unroll
    for (int i = threadIdx.x; i < total; i += 256) {
      unsigned lds_off = (unsigned)(unsigned long long)(&Wl[i * 4]);
      unsigned long long gaddr =
          (unsigned long long)(const void*)(W + (size_t)i * 4);
      asm volatile("global_load_async_to_lds_b128 %0, %1, off"
                   :: "v"(lds_off), "v"(gaddr) : "memory");
    }
    asm volatile("s_wait_asynccnt 0x0" ::: "memory");
  }
  __syncthreads();

  const int wave = threadIdx.x >> 5;
  const int lane = threadIdx.x & 31;
  const int r0 = blockIdx.x * 128 + wave * 16;
  const int m = lane & 15;
  const int khalf = (lane >> 4) << 1;  // lanes 0-15: K+{0,1}; lanes 16-31: K+{2,3}

  int arow = r0 + m;
  if (arow > M - 1) arow = M - 1;  // clamp; stores are bounds-checked below
  const float* __restrict__ Xrow = X + (size_t)arow * K;

  v8f acc[NT];
#pragma unroll
  for (int nt = 0; nt < NT; ++nt) acc[nt] = (v8f)0.0f;

  for (int k0 = 0; k0 < K; k0 += 4) {
    v2f a;
    a.x = Xrow[k0 + khalf];
    a.y = Xrow[k0 + khalf + 1];
#pragma unroll
    for (int nt = 0; nt < NT; ++nt) {
      const int n = nt * 16 + m;
      v2f b;
      b.x = Wl[(k0 + khalf) * NOUT + n];
      b.y = Wl[(k0 + khalf + 1) * NOUT + n];
      acc[nt] = __builtin_amdgcn_wmma_f32_16x16x4_f32(
          false, a, false, b, (short)0, acc[nt], false, false);
    }
  }

  // C/D layout: VGPR j -> rows r0+j (lanes 0-15) / r0+8+j (lanes 16-31)
  const int rhi = (lane >> 4) * 8;
  if (r0 + 16 <= M) {  // fast path: whole tile in bounds (all but last block)
#pragma unroll
    for (int nt = 0; nt < NT; ++nt) {
      const int col = nt * 16 + m;
#pragma unroll
      for (int j = 0; j < 8; ++j) {
        const size_t idx = (size_t)(r0 + rhi + j) * NOUT + col;
        float v = acc[nt][j];
        if (ACC) v += out[idx];
        out[idx] = v;
      }
    }
  } else {
#pragma unroll
    for (int nt = 0; nt < NT; ++nt) {
      const int col = nt * 16 + m;
#pragma unroll
      for (int j = 0; j < 8; ++j) {
        const int row = r0 + rhi + j;
        if (row < M) {
          const size_t idx = (size_t)row * NOUT + col;
          float v = acc[nt][j];
          if (ACC) v += out[idx];
          out[idx] = v;
        }
      }
    }
  }
}

// ---------------------------------------------------------------------------
// Degree / normalization helpers
// ---------------------------------------------------------------------------
__global__ void deg_init(float* __restrict__ deg, int M) {
  int i = blockIdx.x * blockDim.x + threadIdx.x;
  int st = gridDim.x * blockDim.x;
  for (; i < M; i += st) deg[i] = 1.0f;  // self-loop
}

__global__ void deg_count(const int* __restrict__ dst, float* __restrict__ deg, int E) {
  int i = blockIdx.x * blockDim.x + threadIdx.x;
  int st = gridDim.x * blockDim.x;
  for (; i < E; i += st) atomicAdd(&deg[dst[i]], 1.0f);
}

__global__ void deg_rsqrt(float* __restrict__ deg, int M) {
  int i = blockIdx.x * blockDim.x + threadIdx.x;
  int st = gridDim.x * blockDim.x;
  for (; i < M; i += st) deg[i] = rsqrtf(deg[i]);  // deg >= 1 always
}

// ---------------------------------------------------------------------------
// agg[i,c] = h[i,c]*dinv[i]^2 + bias[c]   (self-loop term + bias, no atomics)
// ---------------------------------------------------------------------------
template <int F>
__global__ void agg_init(const float* __restrict__ h, const float* __restrict__ dinv,
                         const float* __restrict__ bias, float* __restrict__ agg, int M) {
  long long n = (long long)M * F;
  long long i = blockIdx.x * (long long)blockDim.x + threadIdx.x;
  long long st = (long long)gridDim.x * blockDim.x;
  for (; i < n; i += st) {
    int r = (int)(i / F);
    int c = (int)(i & (F - 1));
    float di = dinv[r];
    agg[i] = h[i] * (di * di) + bias[c];
  }
}

// ---------------------------------------------------------------------------
// One wave per edge: agg[dst] += h[src] * dinv[src]*dinv[dst].
// F = 32*VEC features; lane owns VEC consecutive floats. h stays L2-resident
// (51 MB << 192 MB L2); prefetch next edge's source row to hide L2 latency.
// ---------------------------------------------------------------------------
template <int VEC>
__global__ __launch_bounds__(256) void edge_scatter(
    const float* __restrict__ h, const int* __restrict__ src,
    const int* __restrict__ dst, const float* __restrict__ dinv,
    float* __restrict__ agg, int E) {
  constexpr int F = VEC * 32;
  const int lane = threadIdx.x & 31;
  int w = (blockIdx.x * blockDim.x + threadIdx.x) >> 5;
  const int nw = (gridDim.x * blockDim.x) >> 5;
  for (; w < E; w += nw) {
    const int wn = w + nw;  // prefetch next edge's gather row
    if (wn < E) {
      const int sn = src[wn];
      __builtin_prefetch(h + (size_t)sn * F + lane * VEC, 0, 0);
    }
    const int s = src[w];
    const int d = dst[w];
    const float sc = dinv[s] * dinv[d];
    const float* __restrict__ hs = h + (size_t)s * F + lane * VEC;
    float* __restrict__ ad = agg + (size_t)d * F + lane * VEC;
    float v[VEC];
#pragma unroll
    for (int j = 0; j < VEC; ++j) v[j] = hs[j];
#pragma unroll
    for (int j = 0; j < VEC; ++j) atomicAdd(ad + j, v[j] * sc);
  }
}

// ---------------------------------------------------------------------------
// BatchNorm over relu(agg): per-channel sum / sumsq, then normalize in place.
// blockDim.x must equal F; thread c owns channel c (coalesced row reads).
// ---------------------------------------------------------------------------
__global__ void zero_buf(float* __restrict__ p, int n) {
  int i = blockIdx.x * blockDim.x + threadIdx.x;
  if (i < n) p[i] = 0.0f;
}

__global__ void bn_stats(const float* __restrict__ agg, float* __restrict__ stats,
                         int M, int F) {
  const int c = threadIdx.x;
  float s = 0.0f, q = 0.0f;
  for (int r = blockIdx.x; r < M; r += gridDim.x) {
    float v = agg[(size_t)r * F + c];
    v = v > 0.0f ? v : 0.0f;  // relu before stats
    s += v;
    q += v * v;
  }
  atomicAdd(&stats[c], s);
  atomicAdd(&stats[F + c], q);
}

template <int F>
__global__ void bn_apply(float* __restrict__ h, const float* __restrict__ stats,
                         const float* __restrict__ g, const float* __restrict__ b,
                         int M) {
  const float inv = 1.0f / (float)M;
  long long n = (long long)M * F;
  long long i = blockIdx.x * (long long)blockDim.x + threadIdx.x;
  long long st = (long long)gridDim.x * blockDim.x;
  for (; i < n; i += st) {
    int c = (int)(i & (F - 1));
    float mean = stats[c] * inv;
    float var = stats[F + c] * inv - mean * mean;
    float v = h[i];
    v = v > 0.0f ? v : 0.0f;
    h[i] = g[c] * (v - mean) * rsqrtf(var + 1e-5f) + b[c];
  }
}

__global__ void relu_inplace(float* __restrict__ p, long long n) {
  long long i = blockIdx.x * (long long)blockDim.x + threadIdx.x;
  long long st = (long long)gridDim.x * blockDim.x;
  for (; i < n; i += st) {
    float v = p[i];
    p[i] = v > 0.0f ? v : 0.0f;
  }
}

// ---------------------------------------------------------------------------
extern "C" void kernel_launch(void* const* d_in, const int* in_sizes, int n_in,
                              void* d_out, int out_size, void* d_ws, size_t ws_size,
                              hipStream_t stream) {
  const float* x     = (const float*)d_in[0];
  const int*   ei    = (const int*)d_in[1];
  const float* W1    = (const float*)d_in[2];
  const float* b1    = (const float*)d_in[3];
  const float* W2    = (const float*)d_in[4];
  const float* b2    = (const float*)d_in[5];
  const float* Wout  = (const float*)d_in[6];
  const float* bout  = (const float*)d_in[7];
  const float* gamma = (const float*)d_in[8];
  const float* beta  = (const float*)d_in[9];

  const int M = in_sizes[0] / 128;  // 100000
  const int E = in_sizes[1] / 2;    // 1600000
  const int* src = ei;
  const int* dst = ei + E;

  float* ws   = (float*)d_ws;
  float* tmp  = ws;                        // M x 128 (also holds final M x 64 P)
  float* h1   = tmp + (size_t)M * 128;     // M x 128
  float* h2   = h1 + (size_t)M * 128;      // M x 128
  float* dinv = h2 + (size_t)M * 128;      // M      (degree, then rsqrt(degree))
  float* stats = dinv + M;                 // 256

  const int TPB = 256;
  const int gElemM = (M + TPB - 1) / TPB;
  const int gElemMF = (int)(((long long)M * 128 + TPB - 1) / TPB);
  const int gElemMO = (int)(((long long)M * 64 + TPB - 1) / TPB);
  const int gGemm = (M + 127) / 128;
  const int gEdge = (E + 7) / 8;  // 8 edge-waves per 256-thread block
  const int gEdgeCnt = (E + TPB - 1) / TPB;

  // --- normalization: deg -> dinv ---
  deg_init<<<gElemM, TPB, 0, stream>>>(dinv, M);
  deg_count<<<gEdgeCnt, TPB, 0, stream>>>(dst, dinv, E);
  deg_rsqrt<<<gElemM, TPB, 0, stream>>>(dinv, M);

  // --- layer 1: h1 = BN(relu(gcn(x, W1, b1))) ---
  gemm128_wmma<128, false><<<gGemm, TPB, 0, stream>>>(x, W1, tmp, M);
  agg_init<128><<<gElemMF, TPB, 0, stream>>>(tmp, dinv, b1, h1, M);
  edge_scatter<4><<<gEdge, TPB, 0, stream>>>(tmp, src, dst, dinv, h1, E);
  zero_buf<<<1, 256, 0, stream>>>(stats, 256);
  bn_stats<<<1024, 128, 0, stream>>>(h1, stats, M, 128);
  bn_apply<128><<<gElemMF, TPB, 0, stream>>>(h1, stats, gamma, beta, M);

  // --- layer 2: h2 = BN(relu(gcn(h1, W2, b2))) ---
  gemm128_wmma<128, false><<<gGemm, TPB, 0, stream>>>(h1, W2, tmp, M);
  agg_init<128><<<gElemMF, TPB, 0, stream>>>(tmp, dinv, b2, h2, M);
  edge_scatter<4><<<gEdge, TPB, 0, stream>>>(tmp, src, dst, dinv, h2, E);
  zero_buf<<<1, 256, 0, stream>>>(stats, 256);
  bn_stats<<<1024, 128, 0, stream>>>(h2, stats, M, 128);
  bn_apply<128><<<gElemMF, TPB, 0, stream>>>(h2, stats, gamma, beta, M);

  // --- output layer: P = [x|h1|h2] @ Wout  (3 accumulating K=128 GEMMs) ---
  gemm128_wmma<64, false><<<gGemm, TPB, 0, stream>>>(x, Wout, tmp, M);
  gemm128_wmma<64, true><<<gGemm, TPB, 0, stream>>>(h1, Wout + 128 * 64, tmp, M);
  gemm128_wmma<64, true><<<gGemm, TPB, 0, stream>>>(h2, Wout + 256 * 64, tmp, M);

  float* out = (float*)d_out;
  agg_init<64><<<gElemMO, TPB, 0, stream>>>(tmp, dinv, bout, out, M);
  edge_scatter<2><<<gEdge, TPB, 0, stream>>>(tmp, src, dst, dinv, out, E);
  relu_inplace<<<gElemMO, TPB, 0, stream>>>(out, (long long)M * 64);
}